// GAT_48129403519137
// MI455X (gfx1250) — compile-verified
//
#include <hip/hip_runtime.h>
#include <math.h>

// ---------------------------------------------------------------------------
// Types for CDNA5 WMMA (wave32). fp32 path: V_WMMA_F32_16X16X4_F32
//   A: 16x4 f32  -> 2 VGPRs/lane (lanes 0-15: K=k,k+1 ; lanes 16-31: K=k+2,k+3)
//   B: 4x16 f32  -> 2 VGPRs/lane (mirrored K halves, N striped over lanes 0-15)
//   C/D: 16x16 f32 -> 8 VGPRs (VGPR r: lanes0-15 M=r, lanes16-31 M=r+8)
// ---------------------------------------------------------------------------
typedef __attribute__((ext_vector_type(2))) float v2f;
typedef __attribute__((ext_vector_type(8))) float v8f;

#define FEATS 128   // F_in = Heads*Hid = F_out = 128 in this model
#define NEG_SLOPE 0.2f

// ---------------------------------------------------------------------------
// fill
// ---------------------------------------------------------------------------
__global__ void gat_fill_f32(float* __restrict__ p, float v, long long n) {
    long long i = (long long)blockIdx.x * blockDim.x + threadIdx.x;
    if (i < n) p[i] = v;
}

// ---------------------------------------------------------------------------
// C[N,128] = A[N,128] @ B[128,128] with fp32 WMMA 16x16x4.
// Block = 256 threads = 8 waves; block handles a 16-row strip of A,
// wave w computes output columns [16w, 16w+16). K loop: 32 WMMA steps.
// A strip staged in LDS with pad-132 stride (conflict-free fragment reads);
// B (64 KB) is prefetched into WGP$/L2 (global_prefetch) and read per-fragment.
// ---------------------------------------------------------------------------
__global__ __launch_bounds__(256) void gat_gemm128_wmma(
    const float* __restrict__ A, const float* __restrict__ B,
    float* __restrict__ C, int N)
{
    __shared__ float sA[16 * 132];          // 16 rows x 128 cols, padded stride

    const int tid  = threadIdx.x;
    const int wave = tid >> 5;
    const int lane = tid & 31;
    const int m0   = blockIdx.x * 16;

    // Warm all 64KB of B into WGP$/L2 while A staging is in flight:
    // 256 threads x 2 prefetches x 128B lines = 64KB.
    __builtin_prefetch(B + (long long)tid * 32, 0, 3);
    __builtin_prefetch(B + (long long)(tid + 256) * 32, 0, 3);

    // Stage A strip (16x128 = 512 float4) into LDS, zero-pad rows >= N.
    for (int i = 0; i < 2; ++i) {
        int idx  = tid + 256 * i;           // float4 index in strip
        int row  = idx >> 5;                // 0..15
        int col4 = idx & 31;                // 0..31
        float4 v = make_float4(0.f, 0.f, 0.f, 0.f);
        int grow = m0 + row;
        if (grow < N) v = ((const float4*)A)[(long long)grow * 32 + col4];
        float* d = &sA[row * 132 + col4 * 4];
        d[0] = v.x; d[1] = v.y; d[2] = v.z; d[3] = v.w;
    }
    __syncthreads();

    const int nBase = wave * 16;
    const int mrow  = lane & 15;            // M index (A) / N index (B,C)
    const int khalf = (lane >> 4) * 2;      // K sub-offset for this lane half
    const int ncol  = nBase + mrow;

    v8f acc = {};                           // zero C
    #pragma unroll 8
    for (int kk = 0; kk < 128; kk += 4) {
        v2f a, b;
        a.x = sA[mrow * 132 + kk + khalf];
        a.y = sA[mrow * 132 + kk + khalf + 1];
        b.x = B[(long long)(kk + khalf) * FEATS + ncol];
        b.y = B[(long long)(kk + khalf + 1) * FEATS + ncol];
        acc = __builtin_amdgcn_wmma_f32_16x16x4_f32(
            /*neg_a=*/false, a, /*neg_b=*/false, b,
            /*c_mod=*/(short)0, acc, /*reuse_a=*/false, /*reuse_b=*/false);
    }

    // D layout: VGPR r -> M = r + 8*(lane>=16), N = lane&15 (+nBase)
    const int mOff = (lane >> 4) * 8;
    if (m0 + 16 <= N) {
        // full tile (always true when N % 16 == 0): straight-line stores
        #pragma unroll
        for (int r = 0; r < 8; ++r)
            C[(long long)(m0 + mOff + r) * FEATS + ncol] = acc[r];
    } else {
        #pragma unroll
        for (int r = 0; r < 8; ++r) {
            int row = m0 + mOff + r;
            if (row < N) C[(long long)row * FEATS + ncol] = acc[r];
        }
    }
}

// ---------------------------------------------------------------------------
// alpha_s[n,h] = <h[n,h,:], a_src[h,:]> ; alpha_d likewise. One thread/(n,h).
// ---------------------------------------------------------------------------
__global__ void gat_attn_dot(const float* __restrict__ h,
                             const float* __restrict__ a_src,
                             const float* __restrict__ a_dst,
                             float* __restrict__ as_, float* __restrict__ ad_,
                             int N, int H, int C)
{
    long long i = (long long)blockIdx.x * blockDim.x + threadIdx.x;
    if (i >= (long long)N * H) return;
    int hh = (int)(i % H);
    long long n = i / H;
    const float* hp = h + n * (long long)(H * C) + (long long)hh * C;
    float s0 = 0.f, s1 = 0.f;
    for (int c = 0; c < C; ++c) {
        float v = hp[c];
        s0 += v * a_src[hh * C + c];
        s1 += v * a_dst[hh * C + c];
    }
    as_[i] = s0;
    ad_[i] = s1;
}

// float atomic max via monotonic int/uint bit trick (init memory to -inf).
__device__ __forceinline__ void gat_atomicMaxF(float* addr, float val) {
    if (val >= 0.f) atomicMax((int*)addr, __float_as_int(val));
    else            atomicMin((unsigned int*)addr, (unsigned int)__float_as_int(val));
}

// ---------------------------------------------------------------------------
// e = leaky_relu(as[src]+ad[dst]); segment max over dst. One thread/(edge,h).
// ---------------------------------------------------------------------------
__global__ void gat_edge_logits_max(const int* __restrict__ src,
                                    const int* __restrict__ dst,
                                    const float* __restrict__ as_,
                                    const float* __restrict__ ad_,
                                    float* __restrict__ e, float* __restrict__ m,
                                    long long E, int H)
{
    long long i = (long long)blockIdx.x * blockDim.x + threadIdx.x;
    if (i >= E * H) return;
    int hh = (int)(i % H);
    long long eidx = i / H;
    int s = src[eidx], d = dst[eidx];
    float v = as_[(long long)s * H + hh] + ad_[(long long)d * H + hh];
    v = (v > 0.f) ? v : NEG_SLOPE * v;
    e[i] = v;
    gat_atomicMaxF(&m[(long long)d * H + hh], v);
}

// ---------------------------------------------------------------------------
// ex = exp(e - m[dst]); segment sum over dst. e overwritten with ex in place.
// ---------------------------------------------------------------------------
__global__ void gat_edge_exp_sum(const int* __restrict__ dst,
                                 const float* __restrict__ m,
                                 float* __restrict__ e, float* __restrict__ s,
                                 long long E, int H)
{
    long long i = (long long)blockIdx.x * blockDim.x + threadIdx.x;
    if (i >= E * H) return;
    int hh = (int)(i % H);
    long long eidx = i / H;
    int d = dst[eidx];
    float ex = expf(e[i] - m[(long long)d * H + hh]);
    e[i] = ex;
    atomicAdd(&s[(long long)d * H + hh], ex);
}

// ---------------------------------------------------------------------------
// out[dst] += h[src] * alpha. One wave per edge; lane owns a float4 of the
// 128 channels (coalesced 512B gather; 128 f32 atomic adds that land in L2
// since the 25.6MB accumulator fits in the 192MB L2).
// ---------------------------------------------------------------------------
__global__ __launch_bounds__(256) void gat_edge_scatter(
    const int* __restrict__ src, const int* __restrict__ dst,
    const float* __restrict__ hfeat, const float* __restrict__ ex,
    const float* __restrict__ ssum, float* __restrict__ out,
    long long E, int H, int C)
{
    const int lane = threadIdx.x & 31;
    long long edge = (long long)blockIdx.x * 8 + (threadIdx.x >> 5);
    if (edge >= E) return;
    int sN = src[edge], dN = dst[edge];
    int ch = lane * 4;                 // 4 consecutive channels; same head (C%4==0, C>=32)
    int hh = ch / C;
    float al = ex[edge * H + hh] / (ssum[(long long)dN * H + hh] + 1e-16f);
    float4 hv = ((const float4*)(hfeat + (long long)sN * FEATS))[lane];
    float* op = out + (long long)dN * FEATS + ch;
    atomicAdd(op + 0, hv.x * al);
    atomicAdd(op + 1, hv.y * al);
    atomicAdd(op + 2, hv.z * al);
    atomicAdd(op + 3, hv.w * al);
}

// out = relu(out + b), b broadcast over rows of 128.
__global__ void gat_bias_relu(float* __restrict__ out, const float* __restrict__ b,
                              long long n)
{
    long long i = (long long)blockIdx.x * blockDim.x + threadIdx.x;
    if (i >= n) return;
    float v = out[i] + b[i & (FEATS - 1)];
    out[i] = v > 0.f ? v : 0.f;
}

// global mean pool: sums[g,c] += x[n,c]; cnt[g] += 1.
__global__ void gat_pool_sum(const float* __restrict__ x, const int* __restrict__ gid,
                             float* __restrict__ sums, float* __restrict__ cnt,
                             long long n)
{
    long long i = (long long)blockIdx.x * blockDim.x + threadIdx.x;
    if (i >= n) return;
    long long node = i >> 7;
    int c = (int)(i & (FEATS - 1));
    int g = gid[node];
    atomicAdd(&sums[(long long)g * FEATS + c], x[i]);
    if (c == 0) atomicAdd(&cnt[g], 1.0f);
}

// FC [G,128]@[128,32] + log_softmax. One wave per graph, lane = class.
__global__ __launch_bounds__(256) void gat_fc_logsoftmax(
    const float* __restrict__ sums, const float* __restrict__ cnt,
    const float* __restrict__ Wfc, const float* __restrict__ bfc,
    float* __restrict__ outp, int G)
{
    const int lane = threadIdx.x & 31;
    int g = blockIdx.x * 8 + (threadIdx.x >> 5);
    if (g >= G) return;
    float inv = 1.0f / fmaxf(cnt[g], 1.0f);
    float logit = bfc[lane];
    for (int k = 0; k < FEATS; ++k)
        logit += (sums[(long long)g * FEATS + k] * inv) * Wfc[k * 32 + lane];
    float mx = logit;
    for (int off = 16; off >= 1; off >>= 1) mx = fmaxf(mx, __shfl_xor(mx, off, 32));
    float ex = expf(logit - mx);
    float sm = ex;
    for (int off = 16; off >= 1; off >>= 1) sm += __shfl_xor(sm, off, 32);
    outp[(long long)g * 32 + lane] = logit - mx - logf(sm);
}

// ---------------------------------------------------------------------------
// Orchestration
// ---------------------------------------------------------------------------
static inline int ceil_div_ll(long long a, long long b) { return (int)((a + b - 1) / b); }

static void fillf(float* p, float v, long long n, hipStream_t s) {
    gat_fill_f32<<<ceil_div_ll(n, 256), 256, 0, s>>>(p, v, n);
}

extern "C" void kernel_launch(void* const* d_in, const int* in_sizes, int n_in,
                              void* d_out, int out_size, void* d_ws, size_t ws_size,
                              hipStream_t stream)
{
    const float* x    = (const float*)d_in[0];
    const int*   ei   = (const int*)d_in[1];   // [2,E] int32
    const int*   gid  = (const int*)d_in[2];
    const float* W1   = (const float*)d_in[4];
    const float* aS1  = (const float*)d_in[5];
    const float* aD1  = (const float*)d_in[6];
    const float* b1   = (const float*)d_in[7];
    const float* W2   = (const float*)d_in[8];
    const float* aS2  = (const float*)d_in[9];
    const float* aD2  = (const float*)d_in[10];
    const float* b2   = (const float*)d_in[11];
    const float* Wfc  = (const float*)d_in[12];
    const float* bfc  = (const float*)d_in[13];
    float* out = (float*)d_out;

    const long long N = in_sizes[0] / FEATS;
    const long long E = in_sizes[1] / 2;
    const int       G = out_size / 32;        // n_classes = 32
    const int* srcI = ei;
    const int* dstI = ei + E;

    // workspace layout (all fp32, base 16B-aligned by harness)
    float* ws = (float*)d_ws;
    const long long NF = N * FEATS;
    float* h1   = ws;            // [N,128]
    float* out1 = h1   + NF;     // [N,128]
    float* h2   = out1 + NF;     // [N,128]
    float* out2 = h2   + NF;     // [N,128]
    float* as1  = out2 + NF;     // [N,4]
    float* ad1  = as1  + 4*N;
    float* m1   = ad1  + 4*N;
    float* s1   = m1   + 4*N;
    float* as2  = s1   + 4*N;    // [N]
    float* ad2  = as2  + N;
    float* m2   = ad2  + N;
    float* s2   = m2   + N;
    float* sums = s2   + N;      // [G,128]
    float* cnt  = sums + (long long)G * FEATS; // [G]
    float* ex1  = cnt  + G;      // [E,4]
    float* ex2  = ex1  + 4*E;    // [E]

    const float NEG_INF = -__builtin_huge_valf();
    const int gemmBlocks = ceil_div_ll(N, 16);

    // ---------------- layer 1 (H=4, C=32) ----------------
    fillf(out1, 0.f, NF, stream);
    fillf(s1,   0.f, 4*N, stream);
    fillf(m1, NEG_INF, 4*N, stream);

    gat_gemm128_wmma<<<gemmBlocks, 256, 0, stream>>>(x, W1, h1, (int)N);
    gat_attn_dot<<<ceil_div_ll(N*4, 256), 256, 0, stream>>>(h1, aS1, aD1, as1, ad1, (int)N, 4, 32);
    gat_edge_logits_max<<<ceil_div_ll(E*4, 256), 256, 0, stream>>>(srcI, dstI, as1, ad1, ex1, m1, E, 4);
    gat_edge_exp_sum<<<ceil_div_ll(E*4, 256), 256, 0, stream>>>(dstI, m1, ex1, s1, E, 4);
    gat_edge_scatter<<<ceil_div_ll(E, 8), 256, 0, stream>>>(srcI, dstI, h1, ex1, s1, out1, E, 4, 32);
    gat_bias_relu<<<ceil_div_ll(NF, 256), 256, 0, stream>>>(out1, b1, NF);

    // ---------------- layer 2 (H=1, C=128; mean over 1 head == identity) ----
    fillf(out2, 0.f, NF, stream);
    fillf(s2,   0.f, N, stream);
    fillf(m2, NEG_INF, N, stream);

    gat_gemm128_wmma<<<gemmBlocks, 256, 0, stream>>>(out1, W2, h2, (int)N);
    gat_attn_dot<<<ceil_div_ll(N, 256), 256, 0, stream>>>(h2, aS2, aD2, as2, ad2, (int)N, 1, 128);
    gat_edge_logits_max<<<ceil_div_ll(E, 256), 256, 0, stream>>>(srcI, dstI, as2, ad2, ex2, m2, E, 1);
    gat_edge_exp_sum<<<ceil_div_ll(E, 256), 256, 0, stream>>>(dstI, m2, ex2, s2, E, 1);
    gat_edge_scatter<<<ceil_div_ll(E, 8), 256, 0, stream>>>(srcI, dstI, h2, ex2, s2, out2, E, 1, 128);
    gat_bias_relu<<<ceil_div_ll(NF, 256), 256, 0, stream>>>(out2, b2, NF);

    // ---------------- pool + FC + log_softmax ----------------
    fillf(sums, 0.f, (long long)G * FEATS, stream);
    fillf(cnt,  0.f, G, stream);
    gat_pool_sum<<<ceil_div_ll(NF, 256), 256, 0, stream>>>(out2, gid, sums, cnt, NF);
    gat_fc_logsoftmax<<<ceil_div_ll(G, 8), 256, 0, stream>>>(sums, cnt, Wfc, bfc, out, G);
}